// TS2VecLoss_61469571940601
// MI455X (gfx1250) — compile-verified
//
#include <hip/hip_runtime.h>
#include <hip/hip_bf16.h>

typedef float v2f __attribute__((ext_vector_type(2)));
typedef float v8f __attribute__((ext_vector_type(8)));

#define CCH 320
#define KP  160    // CCH/2 k-pairs
#define BB  8
#define T0  2048
#define NEGF (-1.0e30f)
#define BM  128
#define BN  64
// pair-interleaved LDS tiles: element (kp, x) holds {v[2kp][x], v[2kp+1][x]} adjacent.
// pair-strides chosen ≡ 16 (mod 32) so lane-halves (kp vs kp+1) hit disjoint bank sets.
#define ASTP 144
#define BSTP 80

__device__ __forceinline__ float wmax(const float* __restrict__ p, int w) {
  float m = p[0];
  for (int j = 1; j < w; ++j) m = fmaxf(m, p[j]);
  return m;
}

__device__ __forceinline__ v8f vzero8() {
  v8f v = {0.f, 0.f, 0.f, 0.f, 0.f, 0.f, 0.f, 0.f};
  return v;
}

__device__ __forceinline__ v8f wmma_f32(v2f a, v2f b, v8f c) {
  // D = A(16x4) * B(4x16) + C(16x16), fp32, wave32
  return __builtin_amdgcn_wmma_f32_16x16x4_f32(false, a, false, b, (short)0, c, false, false);
}

// ---------------------------------------------------------------------------
// Max-pool level d (window 2^lg) directly from the original input (levels 4..11)
// ---------------------------------------------------------------------------
__global__ __launch_bounds__(256) void pool_kernel(
    const float* __restrict__ z1, const float* __restrict__ z2,
    float* __restrict__ o1, float* __restrict__ o2, int Tlev, int lg) {
  int i = blockIdx.x * blockDim.x + threadIdx.x;
  int n = BB * CCH * Tlev;
  if (i >= n) return;
  int bc = i / Tlev;
  int t  = i - bc * Tlev;
  int w  = 1 << lg;
  const float* p1 = z1 + (size_t)bc * T0 + ((size_t)t << lg);
  const float* p2 = z2 + (size_t)bc * T0 + ((size_t)t << lg);
  o1[i] = wmax(p1, w);
  o2[i] = wmax(p2, w);
}

// ---------------------------------------------------------------------------
// Temporal contrastive loss: for fixed b, G = Z Z^T with Z = [2T, 320],
// Z[t,c] = (t<T ? z1[b,c,t] : z2[b,c,t-T]) (optionally pooled on the fly).
// Workgroup owns BM=128 rows, sweeps BN=64-wide column blocks with WMMA,
// fusing a per-row online (max, sum-exp) LSE + positive-pair gather.
// ---------------------------------------------------------------------------
__global__ __launch_bounds__(256) void tempo_kernel(
    const float* __restrict__ z1, const float* __restrict__ z2,
    int Tlev, int lg, float* __restrict__ partials, int slotBase, float scale) {
  extern __shared__ float lds[];
  float* As = lds;                      // [KP][ASTP][2]
  float* Bs = lds + (size_t)KP * ASTP * 2;  // [KP][BSTP][2]
  __shared__ float wsum[8];

  const int tid  = threadIdx.x;
  const int wave = tid >> 5;
  const int lane = tid & 31;
  const int b    = blockIdx.y;
  const int rowBase = blockIdx.x * BM;
  const int N2   = 2 * Tlev;
  const int w    = 1 << lg;
  const size_t Tsrc = (size_t)Tlev << lg;

  // ---- stage A tile: rows [rowBase, rowBase+BM) over all K=320 ----
  for (int idx = tid; idx < CCH * BM; idx += 256) {
    int m = idx & (BM - 1);
    int k = idx >> 7;          // BM == 128
    int row = rowBase + m;
    float v = 0.f;
    if (row < N2) {
      const float* src = (row < Tlev) ? z1 : z2;
      int t = (row < Tlev) ? row : row - Tlev;
      v = wmax(src + ((size_t)(b * CCH + k)) * Tsrc + ((size_t)t << lg), w);
    }
    As[((k >> 1) * ASTP + m) * 2 + (k & 1)] = v;
  }

  const int ml    = lane & 15;
  const int hk    = lane >> 4;         // 0/1: fragment K half (kp vs kp+1)
  const int halfM = hk << 3;           // 0 or 8 (C/D row split)
  const int mBase = wave * 16;
  const int aOff0 = (hk * ASTP + mBase + ml) * 2;
  const int bOff0 = (hk * BSTP + ml) * 2;

  float Mr[8], Sr[8], Pp[8];
#pragma unroll
  for (int r = 0; r < 8; ++r) { Mr[r] = NEGF; Sr[r] = 0.f; Pp[r] = 0.f; }

  const int nCB = (N2 + BN - 1) / BN;
  for (int jb = 0; jb < nCB; ++jb) {
    const int colBase = jb * BN;
    __syncthreads();  // previous block's B reads done
    for (int idx = tid; idx < CCH * BN; idx += 256) {
      int n = idx & (BN - 1);
      int k = idx >> 6;        // BN == 64
      int col = colBase + n;
      float v = 0.f;
      if (col < N2) {
        const float* src = (col < Tlev) ? z1 : z2;
        int t = (col < Tlev) ? col : col - Tlev;
        v = wmax(src + ((size_t)(b * CCH + k)) * Tsrc + ((size_t)t << lg), w);
      }
      Bs[((k >> 1) * BSTP + n) * 2 + (k & 1)] = v;
    }
    __syncthreads();

    v8f acc[4];
#pragma unroll
    for (int jt = 0; jt < 4; ++jt) acc[jt] = vzero8();

    int aOff = aOff0;
    int bOff = bOff0;
    for (int k0 = 0; k0 < CCH; k0 += 4) {
      v2f a = *(const v2f*)(As + aOff);      // single ds_load_b64 per fragment
#pragma unroll
      for (int jt = 0; jt < 4; ++jt) {
        v2f bf = *(const v2f*)(Bs + bOff + jt * 32);
        acc[jt] = wmma_f32(a, bf, acc[jt]);
      }
      aOff += 4 * ASTP;   // kp advances by 2 per chunk
      bOff += 4 * BSTP;
    }

    // ---- fused masked online log-sum-exp over this 16x64 stripe ----
#pragma unroll
    for (int r = 0; r < 8; ++r) {
      int mg  = rowBase + mBase + halfM + r;                 // global row
      int prt = (mg < Tlev) ? (mg + Tlev) : (mg - Tlev);     // positive column
      float vt[4];
      float mloc = NEGF;
#pragma unroll
      for (int jt = 0; jt < 4; ++jt) {
        int col = colBase + jt * 16 + ml;
        float v = acc[jt][r];
        bool valid = (col < N2) && (col != mg);              // drop diagonal / OOB
        Pp[r] += (valid && (col == prt)) ? v : 0.f;          // branchless gather
        float vm = valid ? v : NEGF;
        vt[jt] = vm;
        mloc = fmaxf(mloc, vm);
      }
#pragma unroll
      for (int off = 1; off < 16; off <<= 1)
        mloc = fmaxf(mloc, __shfl_xor(mloc, off, 32));
      float sl = 0.f;
#pragma unroll
      for (int jt = 0; jt < 4; ++jt) sl += __expf(vt[jt] - mloc);
#pragma unroll
      for (int off = 1; off < 16; off <<= 1)
        sl += __shfl_xor(sl, off, 32);
      float nM = fmaxf(Mr[r], mloc);
      Sr[r] = Sr[r] * __expf(Mr[r] - nM) + sl * __expf(mloc - nM);
      Mr[r] = nM;
    }
  }

  // ---- per-row loss, reduce to one partial per workgroup ----
  float lsum = 0.f;
#pragma unroll
  for (int r = 0; r < 8; ++r) {
    float P = Pp[r];
#pragma unroll
    for (int off = 1; off < 16; off <<= 1) P += __shfl_xor(P, off, 32);
    float rl = Mr[r] + __logf(Sr[r]) - P;
    int mg = rowBase + mBase + halfM + r;
    lsum += (mg < N2) ? rl : 0.f;
  }
  lsum += __shfl_xor(lsum, 16, 32);   // combine lane halves
  if (lane == 0) wsum[wave] = lsum;
  __syncthreads();
  if (tid == 0) {
    float s = 0.f;
    for (int w2 = 0; w2 < 8; ++w2) s += wsum[w2];
    partials[slotBase + blockIdx.y * gridDim.x + blockIdx.x] = s * scale;
  }
}

// ---------------------------------------------------------------------------
// Instance contrastive loss: per slice l, Y = [16, 320] (rows 0..7 z1, 8..15 z2),
// G = Y Y^T via WMMA with identical A/B fragments. One wave per slice.
// ---------------------------------------------------------------------------
__global__ __launch_bounds__(256) void inst_kernel(
    const float* __restrict__ z1, const float* __restrict__ z2,
    int Tlev, int lg, float* __restrict__ partials, int slotBase, float scale) {
  __shared__ float wsum[8];
  const int tid  = threadIdx.x;
  const int wave = tid >> 5;
  const int lane = tid & 31;
  const int l    = blockIdx.x * 8 + wave;
  const int ml   = lane & 15;
  const int halfK = (lane >> 4) << 1;
  const int halfM = (lane >> 4) << 3;
  const int w    = 1 << lg;
  const size_t Tsrc = (size_t)Tlev << lg;

  float lsum = 0.f;
  if (l < Tlev) {
    v8f acc = vzero8();
    const float* srcA = (ml < 8) ? z1 : z2;
    const int na = (ml < 8) ? ml : ml - 8;
    const float* baseA = srcA + ((size_t)(na * CCH)) * Tsrc + ((size_t)l << lg);
    for (int k0 = 0; k0 < CCH; k0 += 4) {
      int ka = k0 + halfK;
      v2f a;
      a.x = wmax(baseA + (size_t)ka * Tsrc, w);
      a.y = wmax(baseA + (size_t)(ka + 1) * Tsrc, w);
      acc = wmma_f32(a, a, acc);   // G = Y Y^T : A and B fragments coincide
    }
#pragma unroll
    for (int r = 0; r < 8; ++r) {
      int mg  = halfM + r;
      int prt = mg ^ 8;
      float v  = acc[r];
      float vm = (ml == mg) ? NEGF : v;
      float P  = (ml == prt) ? v : 0.f;
#pragma unroll
      for (int off = 1; off < 16; off <<= 1) P += __shfl_xor(P, off, 32);
      float mloc = vm;
#pragma unroll
      for (int off = 1; off < 16; off <<= 1) mloc = fmaxf(mloc, __shfl_xor(mloc, off, 32));
      float sl = __expf(vm - mloc);
#pragma unroll
      for (int off = 1; off < 16; off <<= 1) sl += __shfl_xor(sl, off, 32);
      lsum += mloc + __logf(sl) - P;
    }
    lsum += __shfl_xor(lsum, 16, 32);
  }
  if (lane == 0) wsum[wave] = lsum;
  __syncthreads();
  if (tid == 0) {
    float s = 0.f;
    for (int w2 = 0; w2 < 8; ++w2) s += wsum[w2];
    partials[slotBase + blockIdx.x] = s * scale;
  }
}

// ---------------------------------------------------------------------------
// Deterministic final reduction of per-workgroup partials -> 3 scalars
// ---------------------------------------------------------------------------
__global__ __launch_bounds__(256) void finalize_kernel(
    const float* __restrict__ tp, int nT,
    const float* __restrict__ ip, int nI, float* __restrict__ out) {
  __shared__ float red[256];
  int tid = threadIdx.x;
  float s = 0.f;
  for (int i = tid; i < nT; i += 256) s += tp[i];
  red[tid] = s; __syncthreads();
  for (int off = 128; off > 0; off >>= 1) {
    if (tid < off) red[tid] += red[tid + off];
    __syncthreads();
  }
  float tempSum = red[0];
  __syncthreads();
  s = 0.f;
  for (int i = tid; i < nI; i += 256) s += ip[i];
  red[tid] = s; __syncthreads();
  for (int off = 128; off > 0; off >>= 1) {
    if (tid < off) red[tid] += red[tid + off];
    __syncthreads();
  }
  if (tid == 0) {
    float inst = red[0]  / 11.0f;
    float temp = tempSum / 11.0f;
    out[0] = 0.5f * inst + 0.5f * temp;  // ALPHA = 0.5
    out[1] = inst;
    out[2] = temp;
  }
}

// ---------------------------------------------------------------------------
extern "C" void kernel_launch(void* const* d_in, const int* in_sizes, int n_in,
                              void* d_out, int out_size, void* d_ws, size_t ws_size,
                              hipStream_t stream) {
  (void)in_sizes; (void)n_in; (void)out_size; (void)ws_size;
  const float* z1 = (const float*)d_in[0];
  const float* z2 = (const float*)d_in[1];
  float* out = (float*)d_out;
  float* ws  = (float*)d_ws;

  // workspace layout: pooled levels 4..11 for z1/z2, then partial-sum slots
  float* pool1[12] = {nullptr};
  float* pool2[12] = {nullptr};
  size_t off = 0;
  for (int d = 4; d <= 11; ++d) {
    int Tlev = T0 >> d;
    pool1[d] = ws + off; off += (size_t)BB * CCH * Tlev;
    pool2[d] = ws + off; off += (size_t)BB * CCH * Tlev;
  }
  float* tempPart = ws + off; off += 1024;
  float* instPart = ws + off; off += 1024;

  // precompute deep pooled levels straight from the inputs (window = 2^d)
  for (int d = 4; d <= 11; ++d) {
    int Tlev = T0 >> d;
    int n = BB * CCH * Tlev;
    pool_kernel<<<(n + 255) / 256, 256, 0, stream>>>(z1, z2, pool1[d], pool2[d], Tlev, d);
  }

  const size_t ldsBytes = (size_t)(KP * ASTP + KP * BSTP) * 2 * sizeof(float); // 286,720 B

  int tBase = 0, iBase = 0;
  for (int d = 0; d <= 11; ++d) {
    int Tlev = T0 >> d;
    const float* s1 = (d < 4) ? z1 : pool1[d];
    const float* s2 = (d < 4) ? z2 : pool2[d];
    int lg = (d < 4) ? d : 0;                    // on-the-fly pooling window
    float scale = 1.0f / (16.0f * (float)Tlev);  // both losses: 1/(2*N*L)

    dim3 tg((2 * Tlev + BM - 1) / BM, BB);
    tempo_kernel<<<tg, 256, ldsBytes, stream>>>(s1, s2, Tlev, lg, tempPart, tBase, scale);
    tBase += (int)tg.x * BB;

    int ig = (Tlev + 7) / 8;
    inst_kernel<<<ig, 256, 0, stream>>>(s1, s2, Tlev, lg, instPart, iBase, scale);
    iBase += ig;
  }

  finalize_kernel<<<1, 256, 0, stream>>>(tempPart, tBase, instPart, iBase, out);
}